// ToMeLayer_54511724921537
// MI455X (gfx1250) — compile-verified
//
#include <hip/hip_runtime.h>
#include <hip/hip_bf16.h>

typedef __attribute__((ext_vector_type(2))) float v2f;
typedef __attribute__((ext_vector_type(8))) float v8f;

#define B_    8
#define T_    8192
#define D_    1024
#define R_    512
#define TKEEP (T_ - R_)   // 7680
#define NSIM  (T_ - 1)    // 8191
#define EPS_  1e-12f

// ---------------------------------------------------------------------------
// Kernel 1: adjacent cosine similarity via V_WMMA_F32_16X16X4_F32.
// One wave handles 16 sims (token block t0..t0+15 vs t0+1..t0+16).
// A = 16x4 f32 chunk of rows t0..t0+15, B = 4x16 chunk of rows t0+1..t0+16
// (transposed); accumulating over K=1024 gives the Gram tile whose diagonal
// is the adjacent dot products. Norms via per-lane FMA + shfl_xor(16).
// ---------------------------------------------------------------------------
__global__ __launch_bounds__(256)
void tome_sim_kernel(const float* __restrict__ x, float* __restrict__ sim) {
    __shared__ float diag[8 * 256];               // 8 waves * 32 lanes * 8 vgprs
    const int lane = threadIdx.x & 31;
    const int wave = threadIdx.x >> 5;
    const int blk  = blockIdx.x * 8 + wave;       // 0 .. 4095
    const int b    = blk >> 9;                    // / 512 blocks-per-batch
    const int t0   = (blk & 511) << 4;            // 16 sims per wave
    const int rowA = t0 + (lane & 15);
    const int rowB = min(rowA + 1, T_ - 1);       // clamp tail (sim not stored)
    const int coff = (lane >> 4) << 1;            // lanes 0-15: K=0,1 ; 16-31: K=2,3

    const float* xb = x + (size_t)b * T_ * D_;
    const float* pa = xb + (size_t)rowA * D_ + coff;
    const float* pb = xb + (size_t)rowB * D_ + coff;

    v8f acc = {0.f, 0.f, 0.f, 0.f, 0.f, 0.f, 0.f, 0.f};
    float na = 0.f, nb = 0.f;                     // per-lane partial norms^2

#pragma unroll 4
    for (int k = 0; k < D_; k += 4) {
        v2f a  = *(const v2f*)(pa + k);
        v2f bb = *(const v2f*)(pb + k);
        // D = A(16x4) * B(4x16) + C ; true-f32 matrix pipe
        acc = __builtin_amdgcn_wmma_f32_16x16x4_f32(
            false, a, false, bb, (short)0, acc, false, false);
        na = fmaf(a.x, a.x, na);   na = fmaf(a.y, a.y, na);
        nb = fmaf(bb.x, bb.x, nb); nb = fmaf(bb.y, bb.y, nb);
    }
    // combine the two column-halves of each row's norm (lanes l and l+16)
    na += __shfl_xor(na, 16);
    nb += __shfl_xor(nb, 16);

    // spill accumulator tile, then pull the diagonal
    float* wl = diag + wave * 256;
#pragma unroll
    for (int r = 0; r < 8; ++r) wl[lane * 8 + r] = acc[r];
    __syncthreads();

    const int i = lane;                           // sims i = 0..15
    if (i < 16) {
        // C/D layout: vgpr r, lanes 0-15 -> (M=r,N=lane); lanes 16-31 -> (M=r+8,N=lane-16)
        int pos = (i < 8) ? (i * 8 + i) : ((i + 16) * 8 + (i - 8));
        float dot = wl[pos];
        float nl = fmaxf(sqrtf(na), EPS_);
        float nr = fmaxf(sqrtf(nb), EPS_);
        int t = t0 + i;
        if (t < NSIM) sim[(size_t)b * T_ + t] = dot / (nl * nr);
    }
}

// ---------------------------------------------------------------------------
// Kernel 2: per-batch full descending sort (bitonic, 8192 keys in 64KB LDS).
// key = (orderable(sim) << 32) | ~index  -> ties pick smaller index (stable
// argsort of -sim). Writes merge_indices (ws int + d_out float) and flags.
// ---------------------------------------------------------------------------
__global__ __launch_bounds__(1024)
void tome_topk_kernel(const float* __restrict__ sim, int* __restrict__ flags,
                      int* __restrict__ mergei, float* __restrict__ out_merge) {
    __shared__ unsigned long long keys[T_];
    const int b = blockIdx.x;
    const int tid = threadIdx.x;

    for (int i = tid; i < T_; i += 1024) flags[b * T_ + i] = 0;

    for (int i = tid; i < T_; i += 1024) {
        unsigned long long key = 0ull;            // padding sorts last
        if (i < NSIM) {
            unsigned u = __float_as_uint(sim[(size_t)b * T_ + i]);
            unsigned ord = (u & 0x80000000u) ? ~u : (u | 0x80000000u);
            key = ((unsigned long long)ord << 32) |
                  (unsigned long long)(0xFFFFFFFFu - (unsigned)i);
        }
        keys[i] = key;
    }
    __syncthreads();

    for (int kk = 2; kk <= T_; kk <<= 1) {
        for (int j = kk >> 1; j > 0; j >>= 1) {
            for (int i = tid; i < T_; i += 1024) {
                int ixj = i ^ j;
                if (ixj > i) {
                    unsigned long long a = keys[i], c = keys[ixj];
                    bool up = ((i & kk) == 0);
                    bool sw = up ? (a < c) : (a > c);   // overall descending
                    if (sw) { keys[i] = c; keys[ixj] = a; }
                }
            }
            __syncthreads();
        }
    }

    for (int i = tid; i < R_; i += 1024) {
        int idx = (int)(0xFFFFFFFFu - (unsigned)(keys[i] & 0xFFFFFFFFull));
        int mi  = idx + 1;
        mergei[b * R_ + i]   = mi;
        out_merge[b * R_ + i] = (float)mi;
        flags[b * T_ + mi]   = 1;
    }
}

// ---------------------------------------------------------------------------
// Kernel 3: keep_indices = compacted sorted complement via block scan.
// ---------------------------------------------------------------------------
__global__ __launch_bounds__(1024)
void tome_keep_kernel(const int* __restrict__ flags, int* __restrict__ keepi,
                      float* __restrict__ out_keep) {
    __shared__ int ssum[1024];
    const int b = blockIdx.x;
    const int tid = threadIdx.x;

    int loc[8];
    int cnt = 0;
#pragma unroll
    for (int q = 0; q < 8; ++q) {
        int t = tid * 8 + q;
        int kf = flags[b * T_ + t] ? 0 : 1;
        loc[q] = kf;
        cnt += kf;
    }
    ssum[tid] = cnt;
    __syncthreads();
    for (int off = 1; off < 1024; off <<= 1) {    // inclusive Hillis-Steele
        int v = (tid >= off) ? ssum[tid - off] : 0;
        __syncthreads();
        ssum[tid] += v;
        __syncthreads();
    }
    int pos = ssum[tid] - cnt;                    // exclusive prefix
#pragma unroll
    for (int q = 0; q < 8; ++q) {
        int t = tid * 8 + q;
        if (loc[q]) {
            keepi[b * TKEEP + pos]    = t;
            out_keep[b * TKEEP + pos] = (float)t;
            ++pos;
        }
    }
}

// ---------------------------------------------------------------------------
// Kernel 4: gather kept rows, averaging with successor when successor is a
// merge index. One block per output row, 256 lanes x float4 = 1024 floats.
// ---------------------------------------------------------------------------
__global__ __launch_bounds__(256)
void tome_merge_kernel(const float* __restrict__ x, const int* __restrict__ flags,
                       const int* __restrict__ keepi, float* __restrict__ out) {
    const int row   = blockIdx.x;                 // 0 .. B_*TKEEP-1
    const int b     = row / TKEEP;
    const int k     = row - b * TKEEP;
    const int token = keepi[b * TKEEP + k];
    const bool avg  = (token + 1 < T_) && (flags[b * T_ + token + 1] != 0);

    const float4* src = (const float4*)(x + ((size_t)b * T_ + token) * D_);
    float4*       dst = (float4*)(out + (size_t)row * D_);
    const int c = threadIdx.x;                    // 256 * 16B = 4KB row

    float4 v = src[c];
    if (avg) {
        float4 w = src[c + D_ / 4];               // next token's row
        v.x = (v.x + w.x) * 0.5f;  v.y = (v.y + w.y) * 0.5f;
        v.z = (v.z + w.z) * 0.5f;  v.w = (v.w + w.w) * 0.5f;
    }
    dst[c] = v;
}

// ---------------------------------------------------------------------------
extern "C" void kernel_launch(void* const* d_in, const int* in_sizes, int n_in,
                              void* d_out, int out_size, void* d_ws, size_t ws_size,
                              hipStream_t stream) {
    const float* x = (const float*)d_in[0];
    float* out = (float*)d_out;

    // workspace layout
    char* ws      = (char*)d_ws;
    float* sim    = (float*)ws;                                   // B_*T_ f32
    int*   flags  = (int*)(ws + (size_t)B_ * T_ * 4);             // B_*T_ i32
    int*   mergei = (int*)(ws + (size_t)2 * B_ * T_ * 4);         // B_*R_ i32
    int*   keepi  = (int*)(ws + (size_t)2 * B_ * T_ * 4 + (size_t)B_ * R_ * 4);

    // d_out layout: merged_x | merge_indices | keep_indices (all f32)
    float* out_merged = out;
    float* out_merge  = out + (size_t)B_ * TKEEP * D_;
    float* out_keep   = out_merge + (size_t)B_ * R_;

    // 1) sim: 4096 waves (8 per WG of 256)
    tome_sim_kernel<<<512, 256, 0, stream>>>(x, sim);
    // 2) per-batch descending sort -> merge indices + flags
    tome_topk_kernel<<<B_, 1024, 0, stream>>>(sim, flags, mergei, out_merge);
    // 3) complement compaction -> keep indices
    tome_keep_kernel<<<B_, 1024, 0, stream>>>(flags, keepi, out_keep);
    // 4) streaming gather/average (bandwidth-bound pass)
    tome_merge_kernel<<<B_ * TKEEP, 256, 0, stream>>>(x, flags, keepi, out_merged);
}